// SoftCrossEntropyLoss_48430051230146
// MI455X (gfx1250) — compile-verified
//
#include <hip/hip_runtime.h>
#include <stdint.h>

// Soft cross-entropy: loss = (1/N) * [ sum_n LSE(x_n) - sum_{n,k} t[n,k]*x[n,k] ]
// N=4096, K=50257, fp32. Pure streaming reduction: 1.65 GB read once
// -> ~71 us floor at 23.3 TB/s. Strategy: TDM async loads (HBM -> LDS,
// TENSORcnt-tracked, triple buffered / 2 tiles in flight per wave) +
// single-pass online logsumexp.

#define K_CLASSES     50257
#define BLOCK_SIZE    256
#define WAVES_PER_BLK 8
#define SLICE         256                       // floats per wave per tile
#define NBUF          3                         // pipeline depth: 2 tiles ahead
#define TILE          (SLICE * WAVES_PER_BLK)   // 2048 floats
#define N_TILES       (K_CLASSES / TILE)        // 24 full tiles = 49152 cols
#define TAIL_START    (N_TILES * TILE)          // 49152 (tail = 1105 cols)
#define NEG_FLT_MAX   (-3.402823466e+38f)

typedef unsigned int u32;
typedef u32 u32x4 __attribute__((ext_vector_type(4)));
typedef int  i32x4 __attribute__((ext_vector_type(4)));
typedef int  i32x8 __attribute__((ext_vector_type(8)));

#if defined(__has_builtin)
#if __has_builtin(__builtin_amdgcn_tensor_load_to_lds) && \
    __has_builtin(__builtin_amdgcn_s_wait_tensorcnt)
#define USE_TDM 1
#endif
#endif

#ifdef USE_TDM
// 1-D TDM transfer: nelem fp32 from gsrc (global) into lds_byte_addr.
// D# layout per CDNA5 ISA ch.8 (group0 128b, group1 256b; groups 2/3 zero
// since the tensor is <=2D -> lowers to the 2-group tensor_load_to_lds form).
__device__ __forceinline__ void tdm_load_1d(u32 lds_byte_addr, const float* gsrc,
                                            u32 nelem) {
    uint64_t ga = (uint64_t)(uintptr_t)gsrc;
    u32x4 g0 = {
        1u,                                            // count=1, user mode
        lds_byte_addr,                                 // lds_addr [63:32]
        (u32)(ga & 0xFFFFFFFFu),                       // global_addr lo
        (u32)((ga >> 32) & 0x01FFFFFFu) | (2u << 30)   // global_addr hi | type=2
    };
    i32x8 g1 = {
        (int)(2u << 16),            // data_size = 4 bytes; workgroup_mask = 0
        (int)((nelem & 0xFFFFu) << 16),   // tensor_dim0[15:0] in bits[63:48]
        (int)((nelem >> 16) | (1u << 16)),// tensor_dim0[31:16], tensor_dim1 = 1
        (int)((nelem & 0xFFFFu) << 16),   // tile_dim0 in bits[127:112]
        1,                           // tile_dim1 = 1, tile_dim2 = 0
        (int)nelem,                  // tensor_dim0_stride lo
        0, 0
    };
    i32x4 z4 = { 0, 0, 0, 0 };
#if __clang_major__ >= 23
    i32x8 z8 = { 0, 0, 0, 0, 0, 0, 0, 0 };
    __builtin_amdgcn_tensor_load_to_lds(g0, g1, z4, z4, z8, 0);
#else
    __builtin_amdgcn_tensor_load_to_lds(g0, g1, z4, z4, 0);
#endif
}
#endif

__device__ __forceinline__ void fetch_slice(float* lds_dst, const float* gsrc,
                                            int lane) {
#ifdef USE_TDM
    (void)lane;
    u32 laddr = (u32)(uintptr_t)lds_dst;
    laddr = (u32)__builtin_amdgcn_readfirstlane((int)laddr);
    tdm_load_1d(laddr, gsrc, (u32)SLICE);
#else
    #pragma unroll
    for (int j = 0; j < SLICE / 32; ++j)
        lds_dst[lane + 32 * j] = gsrc[lane + 32 * j];
#endif
}

// s_wait_tensorcnt needs a compile-time-constant SIMM16 -> template parameter.
template <int CNT>
__device__ __forceinline__ void wait_tensor() {
#ifdef USE_TDM
    __builtin_amdgcn_s_wait_tensorcnt(CNT);
#endif
    __asm__ volatile("" ::: "memory");   // keep LDS reads below the wait
}

// Online logsumexp update + dot accumulation for one element.
__device__ __forceinline__ void accum(float xv, float tv,
                                      float& m, float& s, float& d) {
    float mn = fmaxf(m, xv);
    s = s * __expf(m - mn) + __expf(xv - mn);
    m = mn;
    d = fmaf(tv, xv, d);
}

__global__ __launch_bounds__(BLOCK_SIZE)
void sxent_rows(const float* __restrict__ X, const float* __restrict__ T,
                float* __restrict__ row_out) {
    __shared__ __align__(16) float xbuf[NBUF][WAVES_PER_BLK][SLICE];
    __shared__ __align__(16) float tbuf[NBUF][WAVES_PER_BLK][SLICE];
    __shared__ float red_m[BLOCK_SIZE], red_s[BLOCK_SIZE], red_d[BLOCK_SIZE];

    const int tid  = threadIdx.x;
    const int lane = tid & 31;
    const int wave = __builtin_amdgcn_readfirstlane(tid >> 5);
    const int row  = blockIdx.x;

    const float* xrow = X + (size_t)row * K_CLASSES;
    const float* trow = T + (size_t)row * K_CLASSES;
    const int slice0 = wave * SLICE;

    float m = NEG_FLT_MAX, s = 0.0f, d = 0.0f;

    // Prologue: async-fetch tiles 0 and 1 (each wave fetches only its own
    // slice, so there are no cross-wave barriers in the stream loop).
    fetch_slice(&xbuf[0][wave][0], xrow + slice0, lane);
    fetch_slice(&tbuf[0][wave][0], trow + slice0, lane);
    fetch_slice(&xbuf[1][wave][0], xrow + TILE + slice0, lane);
    fetch_slice(&tbuf[1][wave][0], trow + TILE + slice0, lane);

    int b = 0, nb = 2;                 // buffer of tile i / of tile i+2
    for (int i = 0; i < N_TILES; ++i) {
        if (i + 2 < N_TILES) {
            const int off = (i + 2) * TILE + slice0;
            fetch_slice(&xbuf[nb][wave][0], xrow + off, lane);
            fetch_slice(&tbuf[nb][wave][0], trow + off, lane);
            wait_tensor<4>();          // tile i done; i+1, i+2 in flight
        } else if (i + 1 < N_TILES) {
            wait_tensor<2>();          // drain: only tile i+1 in flight
        } else {
            wait_tensor<0>();
        }
        // Consume this wave's 256-float slice: 8 floats/lane, ds_load_b128.
        const float4* xv4 = (const float4*)&xbuf[b][wave][0] + lane * 2;
        const float4* tv4 = (const float4*)&tbuf[b][wave][0] + lane * 2;
        #pragma unroll
        for (int j = 0; j < 2; ++j) {
            float4 x = xv4[j];
            float4 t = tv4[j];
            accum(x.x, t.x, m, s, d);
            accum(x.y, t.y, m, s, d);
            accum(x.z, t.z, m, s, d);
            accum(x.w, t.w, m, s, d);
        }
        __asm__ volatile("" ::: "memory");  // reads done before buffer reuse
        b  = (b  + 1 == NBUF) ? 0 : b  + 1;
        nb = (nb + 1 == NBUF) ? 0 : nb + 1;
    }

    // Ragged tail: columns [49152, 50257), coalesced non-temporal loads
    // (read-once data; keep it out of the caches).
    for (int c = TAIL_START + tid; c < K_CLASSES; c += BLOCK_SIZE) {
        float xv = __builtin_nontemporal_load(xrow + c);
        float tv = __builtin_nontemporal_load(trow + c);
        accum(xv, tv, m, s, d);
    }

    // Workgroup tree-reduction with the (m,s) logsumexp merge rule.
    red_m[tid] = m; red_s[tid] = s; red_d[tid] = d;
    __syncthreads();
    for (int stride = BLOCK_SIZE / 2; stride > 0; stride >>= 1) {
        if (tid < stride) {
            float m1 = red_m[tid], s1 = red_s[tid];
            float m2 = red_m[tid + stride], s2 = red_s[tid + stride];
            float mn = fmaxf(m1, m2);
            red_s[tid] = s1 * __expf(m1 - mn) + s2 * __expf(m2 - mn);
            red_m[tid] = mn;
            red_d[tid] += red_d[tid + stride];
        }
        __syncthreads();
    }
    if (tid == 0) {
        float lse = red_m[0] + __logf(red_s[0]);   // LSE = m + log(s)
        row_out[row] = lse - red_d[0];             // per-row loss contribution
    }
}

// Deterministic final reduction of per-row partials (no float atomics, so the
// result is bit-stable across graph replays).
__global__ __launch_bounds__(256)
void final_reduce(const float* __restrict__ rows, float* __restrict__ out,
                  int n) {
    __shared__ float sm[256];
    float a = 0.0f;
    for (int i = threadIdx.x; i < n; i += 256) a += rows[i];
    sm[threadIdx.x] = a;
    __syncthreads();
    for (int s = 128; s > 0; s >>= 1) {
        if (threadIdx.x < s) sm[threadIdx.x] += sm[threadIdx.x + s];
        __syncthreads();
    }
    if (threadIdx.x == 0) out[0] = sm[0] / (float)n;
}

extern "C" void kernel_launch(void* const* d_in, const int* in_sizes, int n_in,
                              void* d_out, int out_size, void* d_ws, size_t ws_size,
                              hipStream_t stream) {
    const float* X = (const float*)d_in[0];   // logits, (N, K) fp32
    const float* T = (const float*)d_in[1];   // soft targets, (N, K) fp32
    float* out  = (float*)d_out;              // scalar fp32
    float* rows = (float*)d_ws;               // N per-row partials (16 KB)

    const int N = in_sizes[0] / K_CLASSES;    // 4096

    sxent_rows<<<N, BLOCK_SIZE, 0, stream>>>(X, T, rows);
    final_reduce<<<1, 256, 0, stream>>>(rows, out, N);
}